// MeanIoU_76218489635189
// MI455X (gfx1250) — compile-verified
//
#include <hip/hip_runtime.h>
#include <hip/hip_bf16.h>

// MeanIoU for prediction [16,21,512,512] f32, mask [16,512,512] int.
// Output: 21 per-class IoU + 1 mIoU = 22 floats.
//
// Bandwidth-bound: 352MB logits + 17MB mask read once -> ~16us floor @ 23.3TB/s.
// Strategy: b128 non-temporal streaming loads, register argmax, LDS-histogram
// per block, low-fan-in global atomics, WMMA-based final reduction.

#define NC 21
#define PLANE (512 * 512)

typedef float v4f __attribute__((ext_vector_type(4)));
typedef int   v4i __attribute__((ext_vector_type(4)));
typedef float v2f __attribute__((ext_vector_type(2)));
typedef float v8f __attribute__((ext_vector_type(8)));

// ---------------------------------------------------------------------------
// Zero the 63 (rounded to 64) global counters in d_ws (harness poisons d_ws).
__global__ void miou_zero_kernel(unsigned int* __restrict__ counts) {
    if (threadIdx.x < 64) counts[threadIdx.x] = 0u;
}

// ---------------------------------------------------------------------------
// Main pass: per-pixel argmax over 21 channels + histogram accumulation.
// Each thread handles chunks of 4 consecutive pixels (float4 per channel).
__global__ __launch_bounds__(256) void miou_count_kernel(
    const float* __restrict__ pred,
    const int*   __restrict__ mask,
    unsigned int* __restrict__ counts,   // [0..20]=mask, [21..41]=pred, [42..62]=inter
    int nchunk) {
    __shared__ unsigned int sc[3 * NC];
    const int t = threadIdx.x;
    for (int j = t; j < 3 * NC; j += 256) sc[j] = 0u;
    __syncthreads();

    const int stride = gridDim.x * 256;
    for (int i = blockIdx.x * 256 + t; i < nchunk; i += stride) {
        const size_t p0 = (size_t)i << 2;                 // base pixel index
        const size_t b  = p0 / (size_t)PLANE;             // batch
        const size_t s  = p0 % (size_t)PLANE;             // offset in plane
        const float* base = pred + b * (size_t)(NC * PLANE) + s;

        // channel 0 initializes the running max (argmax first-occurrence)
        v4f m = __builtin_nontemporal_load((const v4f*)base);
        int am0 = 0, am1 = 0, am2 = 0, am3 = 0;
#pragma unroll
        for (int c = 1; c < NC; ++c) {
            v4f v = __builtin_nontemporal_load((const v4f*)(base + (size_t)c * PLANE));
            if (v.x > m.x) { m.x = v.x; am0 = c; }
            if (v.y > m.y) { m.y = v.y; am1 = c; }
            if (v.z > m.z) { m.z = v.z; am2 = c; }
            if (v.w > m.w) { m.w = v.w; am3 = c; }
        }

        const v4i lab = __builtin_nontemporal_load((const v4i*)(mask + p0));

        atomicAdd(&sc[lab.x], 1u);
        atomicAdd(&sc[lab.y], 1u);
        atomicAdd(&sc[lab.z], 1u);
        atomicAdd(&sc[lab.w], 1u);
        atomicAdd(&sc[NC + am0], 1u);
        atomicAdd(&sc[NC + am1], 1u);
        atomicAdd(&sc[NC + am2], 1u);
        atomicAdd(&sc[NC + am3], 1u);
        if (am0 == lab.x) atomicAdd(&sc[2 * NC + lab.x], 1u);
        if (am1 == lab.y) atomicAdd(&sc[2 * NC + lab.y], 1u);
        if (am2 == lab.z) atomicAdd(&sc[2 * NC + lab.z], 1u);
        if (am3 == lab.w) atomicAdd(&sc[2 * NC + lab.w], 1u);
    }

    __syncthreads();
    for (int j = t; j < 3 * NC; j += 256) {
        const unsigned int v = sc[j];
        if (v) atomicAdd(&counts[j], v);
    }
}

// ---------------------------------------------------------------------------
// Finalize (1 wave32): per-class IoU, then mIoU via one V_WMMA_F32_16X16X4_F32:
//   A rows 0-5  = iou[0..23]  (tail zero-padded)
//   A rows 8-13 = present[0..23]
//   B = ones(4x16)  =>  D row sums give sum(iou) and sum(present).
__global__ void miou_finalize_kernel(const unsigned int* __restrict__ counts,
                                     float* __restrict__ out) {
    __shared__ float s_iou[24];
    __shared__ float s_pre[24];
    const int t = threadIdx.x;
    if (t < 24) { s_iou[t] = 0.0f; s_pre[t] = 0.0f; }
    __syncthreads();

    if (t < NC) {
        const unsigned int mc = counts[t];
        const unsigned int pc = counts[NC + t];
        const unsigned int ic = counts[2 * NC + t];
        unsigned int un = mc + pc - ic;
        if (un == 0u) un = 1u;                       // jnp.maximum(union, 1)
        const float pres = (mc > 0u) ? 1.0f : 0.0f;  // class present in mask
        const float iou  = (mc > 0u) ? ((float)ic / (float)un) : 0.0f;
        out[t]   = iou;
        s_iou[t] = iou;
        s_pre[t] = pres;
    }
    __syncthreads();

#if __has_builtin(__builtin_amdgcn_wmma_f32_16x16x4_f32)
    // A-matrix 16x4 f32 layout (wave32): lanes 0-15 -> M=lane, K={0,1};
    // lanes 16-31 -> M=lane-16, K={2,3}.  A[M][K] = arr[M*4+K].
    const int m  = t & 15;
    const int kb = (t < 16) ? 0 : 2;
    v2f a; a.x = 0.0f; a.y = 0.0f;
    if (m < 6) {
        a.x = s_iou[m * 4 + kb];
        a.y = s_iou[m * 4 + kb + 1];
    } else if (m >= 8 && m < 14) {
        const int r = m - 8;
        a.x = s_pre[r * 4 + kb];
        a.y = s_pre[r * 4 + kb + 1];
    }
    v2f bones; bones.x = 1.0f; bones.y = 1.0f;
    v8f c0 = {};
    v8f d = __builtin_amdgcn_wmma_f32_16x16x4_f32(
        /*neg_a=*/false, a, /*neg_b=*/false, bones,
        /*c_mod=*/(short)0, c0, /*reuse_a=*/false, /*reuse_b=*/false);
    // lanes 0-15: d[g] = rowsum(row g) -> rows 0-5 hold iou partials
    // lanes 16-31: d[g] = rowsum(row 8+g) -> rows 8-13 hold present partials
    const float sum6  = d[0] + d[1] + d[2] + d[3] + d[4] + d[5];
    const float spres = __shfl(sum6, 16, 32);        // present-sum lives in lane 16
    if (t == 0) out[NC] = sum6 / spres;              // mIoU
#else
    if (t == 0) {
        float si = 0.0f, sp = 0.0f;
        for (int c = 0; c < NC; ++c) { si += s_iou[c]; sp += s_pre[c]; }
        out[NC] = si / sp;
    }
#endif
}

// ---------------------------------------------------------------------------
extern "C" void kernel_launch(void* const* d_in, const int* in_sizes, int n_in,
                              void* d_out, int out_size, void* d_ws, size_t ws_size,
                              hipStream_t stream) {
    const float* pred  = (const float*)d_in[0];
    const int*   mask  = (const int*)d_in[1];
    float*       out   = (float*)d_out;
    unsigned int* cnts = (unsigned int*)d_ws;

    const int npix   = in_sizes[1];       // 16*512*512 = 4,194,304
    const int nchunk = npix >> 2;         // 4 pixels per chunk

    miou_zero_kernel<<<1, 64, 0, stream>>>(cnts);

    // 1024 blocks * 256 threads, grid-stride over chunks: saturates HBM while
    // keeping the final global-atomic fan-in tiny (63 bins * 1024 blocks).
    miou_count_kernel<<<1024, 256, 0, stream>>>(pred, mask, cnts, nchunk);

    miou_finalize_kernel<<<1, 32, 0, stream>>>(cnts, out);
}